// UserCollaborativeFiltering_22136261443917
// MI455X (gfx1250) — compile-verified
//
#include <hip/hip_runtime.h>

// Problem constants (from reference): T=64, U=142, I=4500, B=16384, K=10
#define T_DIM 64
#define U_DIM 142
#define I_DIM 4500
#define B_DIM 16384
#define K_TOP 10

#define WAVES_PER_BLOCK 8
#define BLOCK_THREADS   (WAVES_PER_BLOCK * 32)
#define VPL 5            // ceil(142/32) candidate users per lane
#define LDS_STRIDE 160   // padded sim-row slot per wave (floats)

__global__ __launch_bounds__(BLOCK_THREADS)
void ucf_topk_kernel(const float* __restrict__ qos,       // [T,U,I]
                     const float* __restrict__ user_avg,  // [T,U]
                     const float* __restrict__ sim,       // [U,U]
                     const int*   __restrict__ user_id,   // [B]
                     const int*   __restrict__ item_id,   // [B]
                     const int*   __restrict__ time_id,   // [B]
                     float*       __restrict__ out)       // [B]
{
    __shared__ float s_sim[WAVES_PER_BLOCK * LDS_STRIDE];

    const int wave = threadIdx.x >> 5;
    const int lane = threadIdx.x & 31;
    const int b    = blockIdx.x * WAVES_PER_BLOCK + wave;   // wave-uniform
    if (b >= B_DIM) return;

    const int u  = user_id[b];
    const int it = item_id[b];
    const int t  = time_id[b];

    // ---- Stage sim[u, :] into this wave's private LDS slot via the CDNA5
    // ---- async-tensor path (ASYNCcnt), overlapping with the qos gathers below.
    float* slot = &s_sim[wave * LDS_STRIDE];
    // Low 32 bits of the flat shared pointer == LDS byte offset (aperture layout).
    const unsigned lds_base = (unsigned)(uintptr_t)slot;
    const float* simrow = sim + (size_t)u * U_DIM;
#pragma unroll
    for (int j = 0; j < VPL; ++j) {
        const int v  = lane + 32 * j;
        const int vc = (v < U_DIM) ? v : (U_DIM - 1);   // clamp: keep EXEC full
        const unsigned ldsa = lds_base + (unsigned)(4 * v);
        const float* g = simrow + vc;
        asm volatile("global_load_async_to_lds_b32 %0, %1, off"
                     :: "v"(ldsa), "v"(g) : "memory");
    }

    // ---- Long-latency strided gather of the item column qos[t, v, it].
    float val[VPL];
    const size_t base = (size_t)t * U_DIM * I_DIM + (size_t)it;
#pragma unroll
    for (int j = 0; j < VPL; ++j) {
        const int v  = lane + 32 * j;
        const int vc = (v < U_DIM) ? v : (U_DIM - 1);
        val[j] = qos[base + (size_t)vc * I_DIM];
    }

    // ---- Consume the async-staged sim row.
    asm volatile("s_wait_asynccnt 0" ::: "memory");

    float m[VPL];
#pragma unroll
    for (int j = 0; j < VPL; ++j) {
        const int v = lane + 32 * j;
        const float sv = slot[v];                 // ds_load_b32
        const bool valid = (v < U_DIM);
        const bool rated = valid && (val[j] > 0.0f);
        // rated -> sim value; unrated (valid) -> 0.0 candidate; invalid -> -inf
        m[j] = rated ? sv : (valid ? 0.0f : -INFINITY);
    }

    const float* avg_row = user_avg + (size_t)t * U_DIM;
    const float avg_u = avg_row[u];

    // ---- Iterative top-10: wave-wide argmax, remove, accumulate.
    float S = 0.0f;   // sum of selected (pre-normalization) weights
    float W = 0.0f;   // sum of w * (r_v - avg_v)
#pragma unroll 1
    for (int k = 0; k < K_TOP; ++k) {
        float best = -INFINITY;
        int   bidx = 0x7fffffff;
#pragma unroll
        for (int j = 0; j < VPL; ++j) {
            const int v = lane + 32 * j;
            if (m[j] > best) { best = m[j]; bidx = v; }   // strict > keeps lowest v on ties
        }
#pragma unroll
        for (int off = 16; off >= 1; off >>= 1) {
            const float ov = __shfl_xor(best, off, 32);
            const int   oi = __shfl_xor(bidx, off, 32);
            if (ov > best || (ov == best && oi < bidx)) { best = ov; bidx = oi; }
        }
        // bidx/best are now wave-uniform. Owner lane = bidx%32, slot = bidx/32.
        const int oj = bidx >> 5;
        const int ol = bidx & 31;

        // Broadcast the owner's qos value (r_v) without regathering.
        float rloc = val[0];
#pragma unroll
        for (int j = 1; j < VPL; ++j) if (oj == j) rloc = val[j];
        const float r = __shfl(rloc, ol, 32);

        // Remove the selected entry.
#pragma unroll
        for (int j = 0; j < VPL; ++j)
            if (oj == j && lane == ol) m[j] = -INFINITY;

        const float av = avg_row[bidx];           // wave-uniform cached load
        S += best;
        W += best * (r - av);
    }

    const float denom = S + 1e-8f;
    float contrib = (denom != 0.0f) ? (W / denom) : 0.0f;
    if (contrib != contrib) contrib = 0.0f;       // nan_to_num on normalized weights

    if (lane == 0) out[b] = avg_u + contrib;
}

extern "C" void kernel_launch(void* const* d_in, const int* in_sizes, int n_in,
                              void* d_out, int out_size, void* d_ws, size_t ws_size,
                              hipStream_t stream)
{
    const float* qos      = (const float*)d_in[0];  // qos_tensor [T,U,I]
    const float* user_avg = (const float*)d_in[1];  // [T,U]
    const float* sim      = (const float*)d_in[2];  // user_sim_agg [U,U]
    const int*   user_id  = (const int*)  d_in[3];  // [B]
    const int*   item_id  = (const int*)  d_in[4];  // [B]
    const int*   time_id  = (const int*)  d_in[5];  // [B]
    // d_in[6] = top_k scalar (always 10)
    float* out = (float*)d_out;

    const int grid = (B_DIM + WAVES_PER_BLOCK - 1) / WAVES_PER_BLOCK;  // 2048
    ucf_topk_kernel<<<grid, BLOCK_THREADS, 0, stream>>>(
        qos, user_avg, sim, user_id, item_id, time_id, out);
}